// GCN_26414048870993
// MI455X (gfx1250) — compile-verified
//
#include <hip/hip_runtime.h>

typedef __attribute__((ext_vector_type(2))) float        v2f;
typedef __attribute__((ext_vector_type(8))) float        v8f;
typedef __attribute__((ext_vector_type(4))) unsigned int v4u;
typedef __attribute__((ext_vector_type(4))) int          v4i;
typedef __attribute__((ext_vector_type(8))) int          v8i;

#define N_NODES    50000
#define FDIM       128
#define MT         5      // M-tiles (16 rows each) per wave; 3125 = 625*5
#define NCOLS      64     // W columns staged per block (4 waves x 16)
#define LDS_STRIDE 65     // 64 + 1 DWORD pad inserted by TDM (bank-conflict free)

// ---------------------------------------------------------------------------
// Kernel 1: zero the aggregation buffer h (in d_ws).
// ---------------------------------------------------------------------------
__global__ void GCN_zero_kernel(float4* __restrict__ p, int n4) {
    int i = blockIdx.x * blockDim.x + threadIdx.x;
    if (i < n4) {
        float4 z = {0.f, 0.f, 0.f, 0.f};
        p[i] = z;
    }
}

// ---------------------------------------------------------------------------
// Kernel 2: edge gather + scatter-add (copy_src + segment_sum).
// One wave32 per edge: lane L handles floats [4L, 4L+4) of the 128-float row.
// global_load_b128 -> 4 x non-returning global_atomic_add_f32.
// feature (25.6 MB) and h (25.6 MB) are both L2-resident (192 MB L2).
// ---------------------------------------------------------------------------
__global__ void GCN_edge_scatter(const float* __restrict__ feat,
                                 const int*   __restrict__ src,
                                 const int*   __restrict__ dst,
                                 float*       __restrict__ h,
                                 int nedges) {
    const int lane   = threadIdx.x & 31;
    const int wave   = (blockIdx.x * blockDim.x + threadIdx.x) >> 5;
    const int nwaves = (gridDim.x * blockDim.x) >> 5;

    for (int e = wave; e < nedges; e += nwaves) {
        const int s = src[e];
        const int d = dst[e];
        const float4 v = *(const float4*)(feat + (size_t)s * FDIM + lane * 4);
        float* hp = h + (size_t)d * FDIM + lane * 4;
        atomicAdd(hp + 0, v.x);
        atomicAdd(hp + 1, v.y);
        atomicAdd(hp + 2, v.z);
        atomicAdd(hp + 3, v.w);
    }
}

// ---------------------------------------------------------------------------
// Kernel 3: out = h @ W + b via V_WMMA_F32_16X16X4_F32 (full f32 precision).
//
//  * TDM stages W[:, n0b:n0b+64] into LDS once per block (tensor_load_to_lds),
//    with pad_enable inserting +1 DWORD per 64-DWORD row -> LDS stride 65,
//    so lane halves (rows kk and kk+2) never collide on the 64 banks.
//  * Each wave register-caches its full B operand (32 K-quads = 64 VGPRs)
//    from LDS, then sweeps MT=5 M-tiles: 160 WMMAs per wave, A via b64 loads.
//
// ISA register layouts (CDNA5 ISA):
//   A (16x4 f32): lanes 0-15 M=lane / 16-31 M=lane-16; VGPR0=K{0|2}, VGPR1=K{1|3}
//   B (4x16 f32): N striped across lanes; VGPR0=K{0|2}, VGPR1=K{1|3}
//   C/D (16x16 f32): VGPR i -> M = i + 8*(lane>=16), N = lane&15
// ---------------------------------------------------------------------------
__global__ void GCN_gemm_wmma(const float* __restrict__ h,
                              const float* __restrict__ W,
                              const float* __restrict__ bias,
                              float*       __restrict__ out) {
    __shared__ float Wlds[FDIM * LDS_STRIDE];   // 128*65*4 = 33280 B

    const int lane = threadIdx.x & 31;
    const int wib  = threadIdx.x >> 5;          // wave in block (0..3)
    const int n0b  = blockIdx.y * NCOLS;        // 0 or 64
    const int half = lane >> 4;                 // 0 | 1
    const int l15  = lane & 15;

    // ---- TDM: async DMA of the 128x64 f32 tile of W into LDS (wave 0 only;
    //      EXEC is ignored by tensor ops, so the whole-wave branch is the guard)
    if (wib == 0) {
        const unsigned long long ga = (unsigned long long)(const void*)(W + n0b);
        const unsigned ldsa = (unsigned)(size_t)(void*)&Wlds[0];

        v4u g0;
        g0.x = 1u;                                    // count=1, user descriptor
        g0.y = ldsa;                                  // lds_addr (bytes)
        g0.z = (unsigned)(ga & 0xFFFFFFFFull);        // global_addr[31:0]
        g0.w = (unsigned)((ga >> 32) & 0x01FFFFFFull) // global_addr[56:32]
             | (2u << 30);                            // type=2 ("image")

        v8i g1;
        g1[0] = (2 << 16)      // data_size = 4 bytes
              | (1 << 20)      // pad_enable
              | (5 << 22)      // pad_interval: 64 DWORDs
              | (0 << 25);     // pad_amount: 1 DWORD
        g1[1] = (FDIM & 0xFFFF) << 16;                // tensor_dim0 lo16 (=128)
        g1[2] = (FDIM & 0xFFFF) << 16;                // dim0 hi16=0 | tensor_dim1 lo16
        g1[3] = (NCOLS & 0xFFFF) << 16;               // dim1 hi16=0 | tile_dim0=64
        g1[4] = FDIM;                                 // tile_dim1=128, tile_dim2=0
        g1[5] = FDIM;                                 // tensor_dim0_stride lo32 (=128)
        g1[6] = 0;                                    // stride0 hi16, stride1 lo16
        g1[7] = 0;                                    // stride1 hi32

        v4i gz = {0, 0, 0, 0};                        // groups 2/3: unused (2-D)
#if defined(__clang_major__) && (__clang_major__ >= 23)
        v8i gz8 = {0, 0, 0, 0, 0, 0, 0, 0};
        __builtin_amdgcn_tensor_load_to_lds(g0, g1, gz, gz, gz8, 0);
#else
        __builtin_amdgcn_tensor_load_to_lds(g0, g1, gz, gz, 0);
#endif
        __builtin_amdgcn_s_wait_tensorcnt(0);
    }
    __syncthreads();

    // ---- Register-cache the wave's B operand for all K (32 quads, 64 VGPRs)
    const int coff = wib * 16 + l15;                  // column within LDS tile
    v2f Breg[FDIM / 4];
    #pragma unroll
    for (int kq = 0; kq < FDIM / 4; ++kq) {
        const int kk = 4 * kq + 2 * half;
        Breg[kq].x = Wlds[(kk + 0) * LDS_STRIDE + coff];
        Breg[kq].y = Wlds[(kk + 1) * LDS_STRIDE + coff];
    }

    const int   n0    = n0b + wib * 16;
    const float bv    = bias[n0 + l15];
    const int   mbase = blockIdx.x * (MT * 16);

    // ---- Sweep MT M-tiles reusing the same B registers
    for (int mt = 0; mt < MT; ++mt) {
        const int m0 = mbase + mt * 16;
        const float* arow = h + (size_t)(m0 + l15) * FDIM;

        v8f acc = {};
        #pragma unroll
        for (int kq = 0; kq < FDIM / 4; ++kq) {
            const int kk = 4 * kq + 2 * half;
            v2f a;
            a.x = arow[kk + 0];
            a.y = arow[kk + 1];
            // (neg_a, A, neg_b, B, c_mod, C, reuse_a, reuse_b)
            acc = __builtin_amdgcn_wmma_f32_16x16x4_f32(
                false, a, false, Breg[kq], (short)0, acc, false, false);
        }

        #pragma unroll
        for (int i = 0; i < 8; ++i) {
            out[(size_t)(m0 + i + 8 * half) * FDIM + n0 + l15] = acc[i] + bv;
        }
    }
}

// ---------------------------------------------------------------------------
// Launcher
// ---------------------------------------------------------------------------
extern "C" void kernel_launch(void* const* d_in, const int* in_sizes, int n_in,
                              void* d_out, int out_size, void* d_ws, size_t ws_size,
                              hipStream_t stream) {
    const float* feature = (const float*)d_in[0];
    const int*   src     = (const int*)d_in[1];
    const int*   dst     = (const int*)d_in[2];
    const float* W       = (const float*)d_in[3];
    const float* b       = (const float*)d_in[4];
    float*       out     = (float*)d_out;
    float*       h       = (float*)d_ws;   // 50000*128 floats = 25.6 MB scratch

    const int nedges = in_sizes[1];

    // 1) h = 0
    const int n4 = N_NODES * FDIM / 4;
    GCN_zero_kernel<<<(n4 + 255) / 256, 256, 0, stream>>>((float4*)h, n4);

    // 2) scatter-add edges (grid-stride, 8 waves per 256-thread block)
    GCN_edge_scatter<<<4096, 256, 0, stream>>>(feature, src, dst, h, nedges);

    // 3) out = h @ W + b   (625 x 2 blocks; 4 waves/block, 5 M-tiles/wave)
    dim3 grid(N_NODES / (16 * MT), FDIM / NCOLS);
    GCN_gemm_wmma<<<grid, 128, 0, stream>>>(h, W, b, out);
}